// LossFreeRouter_30940944400512
// MI455X (gfx1250) — compile-verified
//
#include <hip/hip_runtime.h>

typedef __attribute__((ext_vector_type(2))) float v2f;
typedef __attribute__((ext_vector_type(8))) float v8f;

#define NEXPERT         64
#define TOKENS_PER_WG   128
#define TOKENS_PER_WAVE 16
#define KC              32    // K-chunk staged in LDS (double-buffered)
#define KSTR            36    // padded LDS row stride (floats) -> conflict-free frags
#define SSTR            68    // score-row stride for the softmax/top-k phase

// float offsets inside the shared block
#define OFF_X0   0
#define OFF_X1   (TOKENS_PER_WG * KSTR)                    // 4608
#define OFF_W0   (2 * TOKENS_PER_WG * KSTR)                // 9216
#define OFF_W1   (OFF_W0 + NEXPERT * KSTR)                 // 11520
#define OFF_B    (OFF_W0 + 2 * NEXPERT * KSTR)             // 13824
#define SMEM_F   (OFF_B + NEXPERT)                         // 13888 floats = 55552 B

__global__ __launch_bounds__(256)
void lfr_router_kernel(const float* __restrict__ x,
                       const float* __restrict__ weight,
                       const float* __restrict__ bias,
                       float* __restrict__ out_scores,
                       float* __restrict__ out_w,
                       float* __restrict__ out_i,
                       int*   __restrict__ counts,
                       int H, int TOPK)
{
    __shared__ __align__(16) float smem[SMEM_F];
    __shared__ int sCnt[NEXPERT];

    const int tid  = threadIdx.x;
    const int lane = tid & 31;
    const int wave = tid >> 5;
    const long tokenBase = (long)blockIdx.x * TOKENS_PER_WG;

    if (tid < NEXPERT) { smem[OFF_B + tid] = bias[tid]; sCnt[tid] = 0; }

    // ---- async global->LDS issue for one K-chunk (B128, ASYNCcnt-tracked) ----
    auto issue_chunk = [&](int k0, int xOff, int wOff) {
        // weight chunk: 64 rows x 32 floats = 512 x 16B, 2 per thread
        #pragma unroll
        for (int it = 0; it < (NEXPERT * KC / 4) / 256; ++it) {
            const int idx = it * 256 + tid;
            const int row = idx >> 3;
            const int c4  = (idx & 7) * 4;
            const unsigned long long g =
                (unsigned long long)(uintptr_t)(weight + (long)row * H + k0 + c4);
            const unsigned lo = (unsigned)(uintptr_t)(const void*)&smem[wOff + row * KSTR + c4];
            asm volatile("global_load_async_to_lds_b128 %0, %1, off"
                         :: "v"(lo), "v"(g) : "memory");
        }
        // x chunk: 128 rows x 32 floats = 1024 x 16B, 4 per thread (coalesced)
        #pragma unroll
        for (int it = 0; it < (TOKENS_PER_WG * KC / 4) / 256; ++it) {
            const int idx = it * 256 + tid;
            const int row = idx >> 3;
            const int c4  = (idx & 7) * 4;
            const unsigned long long g =
                (unsigned long long)(uintptr_t)(x + (tokenBase + row) * (long)H + k0 + c4);
            const unsigned lo = (unsigned)(uintptr_t)(const void*)&smem[xOff + row * KSTR + c4];
            asm volatile("global_load_async_to_lds_b128 %0, %1, off"
                         :: "v"(lo), "v"(g) : "memory");
        }
    };

    // C/D 16x16 f32 layout: VGPR r holds M=r (lanes 0-15) / M=r+8 (lanes 16-31), N=lane%16
    v8f acc0 = {}; v8f acc1 = {}; v8f acc2 = {}; v8f acc3 = {};

    // A 16x4 f32: lanes 0-15 -> K+{0,1}, lanes 16-31 -> K+{2,3}; B mirrors on K rows
    const int kSel = (lane < 16) ? 0 : 2;
    const int mIn  = lane & 15;
    const int waveTokBase = wave * TOKENS_PER_WAVE;

    const int nChunks = H / KC;

    // prologue: stream chunk 0 into buffer 0
    issue_chunk(0, OFF_X0, OFF_W0);
    asm volatile("s_wait_asynccnt 0x0" ::: "memory");
    __syncthreads();

    for (int c = 0; c < nChunks; ++c) {
        const int xo = (c & 1) ? OFF_X1 : OFF_X0;
        const int wo = (c & 1) ? OFF_W1 : OFF_W0;
        if (c + 1 < nChunks)   // overlap: stream next chunk into the other buffer
            issue_chunk((c + 1) * KC, (c & 1) ? OFF_X0 : OFF_X1, (c & 1) ? OFF_W0 : OFF_W1);

        // ---- fp32 WMMA over current chunk: 8 k-steps x 4 expert tiles ----
        const float* aBase = &smem[xo + (waveTokBase + mIn) * KSTR + kSel];
        const float* wBase = &smem[wo + mIn * KSTR + kSel];
        #pragma unroll
        for (int kb = 0; kb < KC; kb += 4) {
            v2f a  = *(const v2f*)(aBase + kb);
            v2f b0 = *(const v2f*)(wBase + kb);
            v2f b1 = *(const v2f*)(wBase + 16 * KSTR + kb);
            v2f b2 = *(const v2f*)(wBase + 32 * KSTR + kb);
            v2f b3 = *(const v2f*)(wBase + 48 * KSTR + kb);
            acc0 = __builtin_amdgcn_wmma_f32_16x16x4_f32(false, a, false, b0, (short)0, acc0, false, false);
            acc1 = __builtin_amdgcn_wmma_f32_16x16x4_f32(false, a, false, b1, (short)0, acc1, false, false);
            acc2 = __builtin_amdgcn_wmma_f32_16x16x4_f32(false, a, false, b2, (short)0, acc2, false, false);
            acc3 = __builtin_amdgcn_wmma_f32_16x16x4_f32(false, a, false, b3, (short)0, acc3, false, false);
        }

        if (c + 1 < nChunks)
            asm volatile("s_wait_asynccnt 0x0" ::: "memory");
        __syncthreads();
    }

    // ---- spill raw score tile into (now free) smem: rows of SSTR ----
    {
        const int mHi = (lane < 16) ? 0 : 8;
        #pragma unroll
        for (int r = 0; r < 8; ++r) {
            const int t = waveTokBase + mHi + r;
            smem[t * SSTR +  0 + mIn] = acc0[r];
            smem[t * SSTR + 16 + mIn] = acc1[r];
            smem[t * SSTR + 32 + mIn] = acc2[r];
            smem[t * SSTR + 48 + mIn] = acc3[r];
        }
    }
    __syncthreads();

    // ---- per-token biased softmax + top-k (threads 0..127, one token each) ----
    if (tid < TOKENS_PER_WG) {
        float* row = &smem[tid * SSTR];
        float mx = -__builtin_inff();
        for (int e = 0; e < NEXPERT; ++e) {
            float s = row[e] + smem[OFF_B + e];
            row[e] = s;
            mx = (s > mx) ? s : mx;
        }
        float sum = 0.0f;
        for (int e = 0; e < NEXPERT; ++e) {
            float p = __expf(row[e] - mx);
            row[e] = p;
            sum += p;
        }
        const float inv = 1.0f / sum;
        const long  gt  = tokenBase + tid;
        float* gs = out_scores + gt * NEXPERT;
        for (int e = 0; e < NEXPERT; e += 4) {
            float4 p;
            p.x = row[e]   * inv; p.y = row[e+1] * inv;
            p.z = row[e+2] * inv; p.w = row[e+3] * inv;
            row[e] = p.x; row[e+1] = p.y; row[e+2] = p.z; row[e+3] = p.w;
            *(float4*)(gs + e) = p;
        }
        // iterative argmax top-k (first-occurrence tie-break matches lax.top_k)
        unsigned long long chosen = 0ull;
        for (int j = 0; j < TOPK; ++j) {
            float best = -__builtin_inff();
            int   bi   = 0;
            for (int e = 0; e < NEXPERT; ++e) {
                if ((chosen >> e) & 1ull) continue;
                float p = row[e];
                if (p > best) { best = p; bi = e; }
            }
            chosen |= 1ull << bi;
            out_w[gt * TOPK + j] = best;
            out_i[gt * TOPK + j] = (float)bi;
            atomicAdd(&sCnt[bi], 1);
        }
    }
    __syncthreads();
    if (tid < NEXPERT && sCnt[tid] != 0) atomicAdd(&counts[tid], sCnt[tid]);
}

__global__ void lfr_zero_counts(int* counts, int e_count) {
    if ((int)threadIdx.x < e_count) counts[threadIdx.x] = 0;
}

__global__ void lfr_bias_update(const int* __restrict__ counts,
                                const float* __restrict__ bias,
                                float* __restrict__ out_bias,
                                float tokens_per_expert, int e_count) {
    int e = threadIdx.x;
    if (e < e_count) {
        float v = (float)counts[e] - tokens_per_expert;
        float s = (v > 0.0f) ? 1.0f : ((v < 0.0f) ? -1.0f : 0.0f);
        out_bias[e] = bias[e] + 0.01f * s;
    }
}

extern "C" void kernel_launch(void* const* d_in, const int* in_sizes, int n_in,
                              void* d_out, int out_size, void* d_ws, size_t ws_size,
                              hipStream_t stream) {
    const float* x = (const float*)d_in[0];
    const float* w = (const float*)d_in[1];
    const float* b = (const float*)d_in[2];

    const long E = in_sizes[2];                 // 64
    const long H = in_sizes[1] / E;             // 4096
    const long N = in_sizes[0] / H;             // 32768
    const int  TOPK = (int)(((long)out_size - N * E - E) / (2 * N));  // 8

    float* out        = (float*)d_out;
    float* out_scores = out;
    float* out_w      = out_scores + N * E;
    float* out_i      = out_w + N * (long)TOPK;
    float* out_bias   = out_i + N * (long)TOPK;
    int*   counts     = (int*)d_ws;

    lfr_zero_counts<<<1, 64, 0, stream>>>(counts, (int)E);

    dim3 grid((unsigned)(N / TOKENS_PER_WG));
    lfr_router_kernel<<<grid, 256, 0, stream>>>(x, w, b, out_scores, out_w, out_i,
                                                counts, (int)H, TOPK);

    const float tokens_per_expert = (float)N * (float)TOPK / (float)E;
    lfr_bias_update<<<1, 64, 0, stream>>>(counts, b, out_bias, tokens_per_expert, (int)E);
}